// NuSelectorModelII_53506702573824
// MI455X (gfx1250) — compile-verified
//
#include <hip/hip_runtime.h>
#include <math.h>

// ---------------- constants from the reference ----------------
#define BATCH   4
#define SEQLEN  1024
#define F_IN    512
#define D_MODEL 256
#define D_INNER 1024        // EXPAND(4) * D_MODEL
#define D_CONV  8
#define D_STATE 16
#define DT_RANK 16          // (256+15)//16
#define NPROJ   48          // DT_RANK + 2*D_STATE
#define OUTPUTS 128
#define MTOT    (BATCH*SEQLEN)   // 4096 rows

typedef __attribute__((ext_vector_type(2))) float v2f;
typedef __attribute__((ext_vector_type(8))) float v8f;

// ---------------- fp32 WMMA fragment helpers (V_WMMA_F32_16X16X4_F32) ------
// A (16x4, MxK): lane L -> m = L&15, khalf = L>>4 ; VGPR0 = A[m][2*khalf], VGPR1 = A[m][2*khalf+1]
__device__ __forceinline__ v2f load_a_frag(const float* A, int lda, int row0, int k0, int lane) {
    const int m = row0 + (lane & 15);
    const int k = k0 + ((lane >> 4) << 1);
    const float* p = A + (size_t)m * lda + k;
    v2f a; a.x = p[0]; a.y = p[1];
    return a;
}
// B (4x16, KxN): lane L -> n = L&15, khalf = L>>4 ; VGPR0 = B[2*khalf][n], VGPR1 = B[2*khalf+1][n]
__device__ __forceinline__ v2f load_b_frag(const float* B, int ldb, int k0, int col0, int lane) {
    const int n = col0 + (lane & 15);
    const int k = k0 + ((lane >> 4) << 1);
    v2f b; b.x = B[(size_t)k * ldb + n]; b.y = B[(size_t)(k + 1) * ldb + n];
    return b;
}

// ---------------- generic fp32 WMMA GEMM: C = act(A@B + bias) ----------------
// act: 0 = identity, 1 = sigmoid
// Block: 256 threads = 8 waves stacked in M; each wave owns a 16 x (16*NTILE)
// strip. Per K-step: batched load phase (A + NTILE B frags -> memory clause),
// then NTILE back-to-back WMMAs (XDL-friendly issue pattern).
template<int NTILE>
__global__ void wmma_gemm_f32(const float* __restrict__ A, const float* __restrict__ B,
                              const float* __restrict__ bias, float* __restrict__ C,
                              int M, int N, int K, int act) {
    const int lane = threadIdx.x & 31;
    const int wave = threadIdx.x >> 5;                 // 0..7
    const int row0 = blockIdx.y * 128 + wave * 16;
    const int col0 = blockIdx.x * (16 * NTILE);
    if (row0 >= M || col0 >= N) return;                // wave-uniform: EXEC stays all-ones

    v8f zero = {};
    v8f acc[NTILE];
    #pragma unroll
    for (int i = 0; i < NTILE; ++i) acc[i] = zero;

    #pragma unroll 4
    for (int k0 = 0; k0 < K; k0 += 4) {
        // ---- load phase: issue all fragment loads for this K-step first ----
        const v2f a = load_a_frag(A, K, row0, k0, lane);
        v2f bfr[NTILE];
        #pragma unroll
        for (int i = 0; i < NTILE; ++i)
            bfr[i] = load_b_frag(B, N, k0, col0 + 16 * i, lane);
        // ---- compute phase: NTILE back-to-back WMMAs reusing the A frag ----
        #pragma unroll
        for (int i = 0; i < NTILE; ++i)
            acc[i] = __builtin_amdgcn_wmma_f32_16x16x4_f32(false, a, false, bfr[i],
                                                           (short)0, acc[i], false, false);
    }

    const int mbase = row0 + ((lane >> 4) << 3);
    #pragma unroll
    for (int i = 0; i < NTILE; ++i) {
        const int n = col0 + 16 * i + (lane & 15);
        const float bv = bias ? bias[n] : 0.0f;
        #pragma unroll
        for (int r = 0; r < 8; ++r) {
            float v = acc[i][r] + bv;
            if (act == 1) v = 1.0f / (1.0f + __expf(-v));
            C[(size_t)(mbase + r) * N + n] = v;
        }
    }
}

// ---------------- depthwise causal conv (D_CONV=8) + SiLU --------------------
// xz: (MTOT, 2*D_INNER) row-major; xc channels are cols [0, D_INNER)
__global__ void conv_silu_kernel(const float* __restrict__ xz,
                                 const float* __restrict__ conv_w,
                                 const float* __restrict__ conv_b,
                                 float* __restrict__ xc) {
    const int idx = blockIdx.x * blockDim.x + threadIdx.x;   // over BATCH*SEQLEN*D_INNER
    if (idx >= BATCH * SEQLEN * D_INNER) return;
    const int d = idx % D_INNER;
    const int t = (idx / D_INNER) % SEQLEN;
    const int b = idx / (D_INNER * SEQLEN);
    const float* xrow = xz + (size_t)b * SEQLEN * (2 * D_INNER);
    float s = conv_b[d];
    #pragma unroll
    for (int j = 0; j < D_CONV; ++j) {
        const int tt = t - (D_CONV - 1) + j;
        if (tt >= 0) s += xrow[(size_t)tt * (2 * D_INNER) + d] * conv_w[d * D_CONV + j];
    }
    xc[idx] = s / (1.0f + __expf(-s));   // silu(s) = s * sigmoid(s)
}

// ---------------- dt = softplus(dt_r @ W_dt + b_dt), K = DT_RANK = 16 --------
__global__ void dt_kernel(const float* __restrict__ dbc, const float* __restrict__ W_dt,
                          const float* __restrict__ b_dt, float* __restrict__ dt) {
    const int idx = blockIdx.x * blockDim.x + threadIdx.x;   // over MTOT*D_INNER
    if (idx >= MTOT * D_INNER) return;
    const int d = idx % D_INNER;
    const int r = idx / D_INNER;
    const float* dr = dbc + (size_t)r * NPROJ;               // dt_r = first 16 cols
    float s = b_dt[d];
    #pragma unroll
    for (int j = 0; j < DT_RANK; ++j) s += dr[j] * W_dt[j * D_INNER + d];
    dt[idx] = (s > 20.0f) ? s : log1pf(__expf(s));
}

// ---------------- selective scan; only y at t = L-1 is needed ----------------
// One lane per (batch, channel, state): serial chain is 1 exp + 1 fma per step.
// 16-lane groups (wave32) reduce y via shfl_xor at the final step.
__global__ void scan_kernel(const float* __restrict__ dt, const float* __restrict__ dbc,
                            const float* __restrict__ xc, const float* __restrict__ xz,
                            const float* __restrict__ A_log, const float* __restrict__ D_param,
                            float* __restrict__ y_last) {
    const int gid = blockIdx.x * blockDim.x + threadIdx.x;   // over BATCH*D_INNER*D_STATE
    if (gid >= BATCH * D_INNER * D_STATE) return;
    const int s  = gid & (D_STATE - 1);
    const int cd = gid >> 4;                 // (b, d) group
    const int d  = cd & (D_INNER - 1);
    const int b  = cd >> 10;                 // D_INNER = 1024

    const float Acoef = -__expf(A_log[d * D_STATE + s]);
    float h = 0.0f;

    const float* dtb  = dt  + (size_t)b * SEQLEN * D_INNER + d;
    const float* xcb  = xc  + (size_t)b * SEQLEN * D_INNER + d;
    const float* dbcb = dbc + (size_t)b * SEQLEN * NPROJ;

    #pragma unroll 4
    for (int t = 0; t < SEQLEN; ++t) {
        const float dtv = dtb[(size_t)t * D_INNER];                    // broadcast (16 lanes)
        const float xv  = xcb[(size_t)t * D_INNER];                    // broadcast
        const float Bs  = dbcb[(size_t)t * NPROJ + DT_RANK + s];       // contiguous in s
        h = __expf(dtv * Acoef) * h + dtv * xv * Bs;
    }

    float y = h * dbcb[(size_t)(SEQLEN - 1) * NPROJ + DT_RANK + D_STATE + s];
    #pragma unroll
    for (int off = 8; off >= 1; off >>= 1)   // masks < 16 keep reduction inside state group
        y += __shfl_xor(y, off, 32);

    if (s == 0) {
        y += xcb[(size_t)(SEQLEN - 1) * D_INNER] * D_param[d];
        const float zl = xz[((size_t)b * SEQLEN + (SEQLEN - 1)) * (2 * D_INNER) + D_INNER + d];
        y_last[(size_t)b * D_INNER + d] = y * (zl / (1.0f + __expf(-zl)));
    }
}

// ---------------- head: (y_last @ W_out) @ W_head + b_head -> softmax --------
// One block per batch. mid = y(1024) @ W_out(1024x256); logits = mid @ W_head(256x128).
__global__ void head_kernel(const float* __restrict__ y_last, const float* __restrict__ W_out,
                            const float* __restrict__ W_head, const float* __restrict__ b_head,
                            float* __restrict__ out) {
    __shared__ float mid[D_MODEL];
    __shared__ float logits[OUTPUTS];
    __shared__ float red[2];
    const int b   = blockIdx.x;
    const int tid = threadIdx.x;
    const float* y = y_last + (size_t)b * D_INNER;

    if (tid < D_MODEL) {
        float s = 0.0f;
        for (int d0 = 0; d0 < D_INNER; ++d0) s += y[d0] * W_out[(size_t)d0 * D_MODEL + tid];
        mid[tid] = s;
    }
    __syncthreads();
    if (tid < OUTPUTS) {
        float s = b_head[tid];
        for (int j = 0; j < D_MODEL; ++j) s += mid[j] * W_head[(size_t)j * OUTPUTS + tid];
        logits[tid] = s;
    }
    __syncthreads();
    if (tid == 0) {
        float mx = logits[0];
        for (int k = 1; k < OUTPUTS; ++k) mx = fmaxf(mx, logits[k]);
        float sum = 0.0f;
        for (int k = 0; k < OUTPUTS; ++k) sum += __expf(logits[k] - mx);
        red[0] = mx; red[1] = sum;
    }
    __syncthreads();
    if (tid < OUTPUTS)
        out[(size_t)b * OUTPUTS + tid] = __expf(logits[tid] - red[0]) / red[1];
}

// ---------------- host-side orchestration ------------------------------------
extern "C" void kernel_launch(void* const* d_in, const int* in_sizes, int n_in,
                              void* d_out, int out_size, void* d_ws, size_t ws_size,
                              hipStream_t stream) {
    const float* x        = (const float*)d_in[0];   // (4,1024,512)
    const float* W_in     = (const float*)d_in[1];   // (512,256)
    const float* b_in     = (const float*)d_in[2];   // (256,)
    const float* W_inproj = (const float*)d_in[3];   // (256,2048)
    const float* conv_w   = (const float*)d_in[4];   // (1024,8)
    const float* conv_b   = (const float*)d_in[5];   // (1024,)
    const float* W_xproj  = (const float*)d_in[6];   // (1024,48)
    const float* W_dt     = (const float*)d_in[7];   // (16,1024)
    const float* b_dt     = (const float*)d_in[8];   // (1024,)
    const float* A_log    = (const float*)d_in[9];   // (1024,16)
    const float* D_param  = (const float*)d_in[10];  // (1024,)
    const float* W_out    = (const float*)d_in[11];  // (1024,256)
    const float* W_head   = (const float*)d_in[12];  // (256,128)
    const float* b_head   = (const float*)d_in[13];  // (128,)
    float* out = (float*)d_out;                      // (4,128)

    // workspace layout (floats)
    float* ws = (float*)d_ws;
    float* H     = ws;                                   // 4096*256
    float* XZ    = H    + (size_t)MTOT * D_MODEL;        // 4096*2048
    float* XC    = XZ   + (size_t)MTOT * 2 * D_INNER;    // 4096*1024
    float* DBC   = XC   + (size_t)MTOT * D_INNER;        // 4096*48
    float* DT    = DBC  + (size_t)MTOT * NPROJ;          // 4096*1024
    float* YLAST = DT   + (size_t)MTOT * D_INNER;        // 4*1024

    dim3 blk(256);

    // 1) H = sigmoid(x @ W_in + b_in)   [4096 x 256, K=512]
    wmma_gemm_f32<4><<<dim3(D_MODEL / 64, MTOT / 128), blk, 0, stream>>>(
        x, W_in, b_in, H, MTOT, D_MODEL, F_IN, 1);

    // 2) XZ = H @ W_inproj              [4096 x 2048, K=256]
    wmma_gemm_f32<4><<<dim3((2 * D_INNER) / 64, MTOT / 128), blk, 0, stream>>>(
        H, W_inproj, nullptr, XZ, MTOT, 2 * D_INNER, D_MODEL, 0);

    // 3) XC = silu(causal depthwise conv(XZ[:, :1024]) + conv_b)
    {
        const int n = BATCH * SEQLEN * D_INNER;
        conv_silu_kernel<<<(n + 255) / 256, blk, 0, stream>>>(XZ, conv_w, conv_b, XC);
    }

    // 4) DBC = XC @ W_xproj             [4096 x 48, K=1024]  (3 x 16-col tiles/wave)
    wmma_gemm_f32<3><<<dim3(1, MTOT / 128), blk, 0, stream>>>(
        XC, W_xproj, nullptr, DBC, MTOT, NPROJ, D_INNER, 0);

    // 5) DT = softplus(dt_r @ W_dt + b_dt)
    {
        const int n = MTOT * D_INNER;
        dt_kernel<<<(n + 255) / 256, blk, 0, stream>>>(DBC, W_dt, b_dt, DT);
    }

    // 6) selective scan (state-parallel) -> gated y at t = L-1 only
    {
        const int n = BATCH * D_INNER * D_STATE;
        scan_kernel<<<(n + 255) / 256, blk, 0, stream>>>(DT, DBC, XC, XZ, A_log, D_param, YLAST);
    }

    // 7) head + softmax on last token
    head_kernel<<<dim3(BATCH), blk, 0, stream>>>(YLAST, W_out, W_head, b_head, out);
}